// SemanticModule_52493090291996
// MI455X (gfx1250) — compile-verified
//
#include <hip/hip_runtime.h>

typedef float v2f __attribute__((ext_vector_type(2)));
typedef float v8f __attribute__((ext_vector_type(8)));

#define NN 100000
#define NE 800000

// ---------------- elementwise helpers ----------------
__global__ void zero_kernel(float* __restrict__ p, long n) {
    long i = (long)blockIdx.x * blockDim.x + threadIdx.x;
    long s = (long)gridDim.x * blockDim.x;
    for (; i < n; i += s) p[i] = 0.0f;
}

// pad x (NN x 6) -> xp (NN x 8), zero cols 6..7
__global__ void pad_x_kernel(const float* __restrict__ x, float* __restrict__ xp) {
    long i = (long)blockIdx.x * blockDim.x + threadIdx.x;
    long s = (long)gridDim.x * blockDim.x;
    for (; i < (long)NN * 8; i += s) {
        int row = (int)(i >> 3);
        int col = (int)(i & 7);
        xp[i] = (col < 6) ? x[(long)row * 6 + col] : 0.0f;
    }
}

// pad w (2 x 6 x 16) -> wp (2 x 8 x 16), zero rows 6..7
__global__ void pad_w_kernel(const float* __restrict__ w, float* __restrict__ wp) {
    int i = blockIdx.x * blockDim.x + threadIdx.x;   // 0..255
    if (i >= 2 * 8 * 16) return;
    int r = i >> 7;
    int k = (i >> 4) & 7;
    int n = i & 15;
    wp[i] = (k < 6) ? w[r * 96 + k * 16 + n] : 0.0f;
}

// ---------------- irregular phase: gather/scatter-add ----------------
// one thread per (edge, feature); consecutive lanes share an edge -> coalesced
// row reads and coalesced atomic destinations within a feature row.
__global__ void scatter_add_kernel(const float* __restrict__ x,
                                   const int* __restrict__ src,
                                   const int* __restrict__ dst,
                                   float* __restrict__ msg, int F) {
    long total = (long)NE * F;
    long i = (long)blockIdx.x * blockDim.x + threadIdx.x;
    long s = (long)gridDim.x * blockDim.x;
    for (; i < total; i += s) {
        int e = (int)(i / F);
        int f = (int)(i - (long)e * F);
        atomicAdd(&msg[(long)dst[e] * F + f], x[(long)src[e] * F + f]);
    }
}

__global__ void scatter_count_kernel(const int* __restrict__ dst,
                                     float* __restrict__ cnt) {
    long i = (long)blockIdx.x * blockDim.x + threadIdx.x;
    long s = (long)gridDim.x * blockDim.x;
    for (; i < NE; i += s) atomicAdd(&cnt[dst[i]], 1.0f);
}

__global__ void mean_div_kernel(float* __restrict__ msg,
                                const float* __restrict__ cnt, int F) {
    long total = (long)NN * F;
    long i = (long)blockIdx.x * blockDim.x + threadIdx.x;
    long s = (long)gridDim.x * blockDim.x;
    for (; i < total; i += s) {
        int row = (int)(i / F);
        msg[i] = msg[i] / fmaxf(cnt[row], 1.0f);
    }
}

// ------- dense phase: D += A1@W1 [+ A2@W2] (+ bias), optional fused ReLU ------
// f32-exact WMMA (V_WMMA_F32_16X16X4_F32). One wave owns a 32x16 output slab
// (two 16x16 tiles sharing the B fragment): two independent accumulators give
// the WMMA pipe dependency-free back-to-back issues, and halve W fetches.
// K,N are compile-time multiples of 4/16 -> fully unrolled, no per-lane guards,
// no EXEC divergence (A2 test is a wave-uniform scalar branch). A pairs load as
// aligned 8-byte v2f (global_load_b64). 100000 = 3125*32 rows, exact.
template <int K, int N, bool RELU>
__global__ __launch_bounds__(256)
void gemm2_wmma(const float* __restrict__ A1, const float* __restrict__ W1,
                const float* __restrict__ A2, const float* __restrict__ W2,
                const float* __restrict__ bias, float* __restrict__ D,
                int nRows) {
    const int lane = threadIdx.x & 31;
    const int wave = threadIdx.x >> 5;
    constexpr int tilesN = N / 16;
    const int totalTiles = (nRows >> 5) * tilesN;      // 32 rows per wave
    const int tile = blockIdx.x * 8 + wave;            // wave-uniform
    if (tile >= totalTiles) return;                    // wave-uniform branch

    const int row0 = (tile / tilesN) << 5;             // 32-row slab base
    const int col  = ((tile % tilesN) << 4) + (lane & 15);
    const int rsel = lane >> 4;                        // 0: lanes 0-15, 1: 16-31
    const int aRow = row0 + (lane & 15);

    // C operands preloaded from D -> WMMA chain computes D += ...
    float* dp0 = D + (long)(row0 + 8 * rsel) * N + col;
    float* dp1 = dp0 + 16 * N;
    v8f c0, c1;
#pragma unroll
    for (int v = 0; v < 8; ++v) { c0[v] = dp0[(long)v * N]; c1[v] = dp1[(long)v * N]; }

    // pass 1: A1 @ W1
    {
        const float* ap0 = A1 + (long)aRow * K + 2 * rsel;
        const float* ap1 = ap0 + 16 * K;
        const float* bp  = W1 + (long)(2 * rsel) * N + col;
#pragma unroll
        for (int kb = 0; kb < K; kb += 4) {
            v2f b; b.x = bp[0]; b.y = bp[N];
            v2f a0 = *(const v2f*)ap0;
            v2f a1 = *(const v2f*)ap1;
            c0 = __builtin_amdgcn_wmma_f32_16x16x4_f32(false, a0, false, b, (short)0, c0, false, false);
            c1 = __builtin_amdgcn_wmma_f32_16x16x4_f32(false, a1, false, b, (short)0, c1, false, false);
            ap0 += 4; ap1 += 4; bp += 4 * N;
        }
    }
    // pass 2 (optional): A2 @ W2 — uniform scalar test, no lane divergence
    if (A2 != nullptr) {
        const float* ap0 = A2 + (long)aRow * K + 2 * rsel;
        const float* ap1 = ap0 + 16 * K;
        const float* bp  = W2 + (long)(2 * rsel) * N + col;
#pragma unroll
        for (int kb = 0; kb < K; kb += 4) {
            v2f b; b.x = bp[0]; b.y = bp[N];
            v2f a0 = *(const v2f*)ap0;
            v2f a1 = *(const v2f*)ap1;
            c0 = __builtin_amdgcn_wmma_f32_16x16x4_f32(false, a0, false, b, (short)0, c0, false, false);
            c1 = __builtin_amdgcn_wmma_f32_16x16x4_f32(false, a1, false, b, (short)0, c1, false, false);
            ap0 += 4; ap1 += 4; bp += 4 * N;
        }
    }

    const float bv = (bias != nullptr) ? bias[col] : 0.0f;
#pragma unroll
    for (int v = 0; v < 8; ++v) {
        float o0 = c0[v] + bv;
        float o1 = c1[v] + bv;
        if (RELU) { o0 = fmaxf(o0, 0.0f); o1 = fmaxf(o1, 0.0f); }
        dp0[(long)v * N] = o0;
        dp1[(long)v * N] = o1;
    }
}

// ---------------- host-side orchestration ----------------
static inline void launch_zero(float* p, long n, hipStream_t s) {
    int blocks = (int)((n + 255) / 256);
    zero_kernel<<<blocks, 256, 0, s>>>(p, n);
}
template <int K, int N, bool RELU>
static inline void launch_gemm2(const float* A1, const float* W1,
                                const float* A2, const float* W2,
                                const float* bias, float* D, int n, hipStream_t s) {
    int totalTiles = (n >> 5) * (N >> 4);
    int blocks = (totalTiles + 7) / 8;                 // 8 waves / 256-thread block
    gemm2_wmma<K, N, RELU><<<blocks, 256, 0, s>>>(A1, W1, A2, W2, bias, D, n);
}
static inline void launch_scatter(const float* x, const int* ei, float* msg,
                                  int F, hipStream_t s) {
    long total = (long)NE * F;
    int blocks = (int)((total + 255) / 256);
    scatter_add_kernel<<<blocks, 256, 0, s>>>(x, ei, ei + NE, msg, F);
}

extern "C" void kernel_launch(void* const* d_in, const int* in_sizes, int n_in,
                              void* d_out, int out_size, void* d_ws, size_t ws_size,
                              hipStream_t stream) {
    const float* x   = (const float*)d_in[0];
    // edges: d_in[1..9] = connected_to, ordered_next, represents, represented_by,
    //        neighboring_vertical, neighboring_horizontal, contains, order, perpendicular
    const float* W0n = (const float*)d_in[10];
    const float* W0r = (const float*)d_in[11];
    const float* b0  = (const float*)d_in[12];
    const float* W1n = (const float*)d_in[13];
    const float* W1r = (const float*)d_in[14];
    const float* b1  = (const float*)d_in[15];
    const float* P1  = (const float*)d_in[16];
    const float* pb1 = (const float*)d_in[17];
    const float* W2n = (const float*)d_in[18];
    const float* W2r = (const float*)d_in[19];
    const float* b2  = (const float*)d_in[20];
    const float* P2  = (const float*)d_in[21];
    const float* pb2 = (const float*)d_in[22];
    float* out = (float*)d_out;

    // workspace layout (floats):
    //   msg[NN*32] | cnt[NN] | h0[NN*16] | h1[NN*32] | xpad[NN*8]
    //   | w0n_pad[2*8*16] | w0r_pad[2*8*16]
    float* msg  = (float*)d_ws;
    float* cnt  = msg  + (long)NN * 32;
    float* h0   = cnt  + NN;
    float* h1   = h0   + (long)NN * 16;
    float* xpad = h1   + (long)NN * 32;
    float* w0np = xpad + (long)NN * 8;
    float* w0rp = w0np + 2 * 8 * 16;

    // ---- prep: pad K=6 operands to K=8 (zero-filled)
    pad_x_kernel<<<(NN * 8 + 255) / 256, 256, 0, stream>>>(x, xpad);
    pad_w_kernel<<<1, 256, 0, stream>>>(W0n, w0np);
    pad_w_kernel<<<1, 256, 0, stream>>>(W0r, w0rp);

    // ---- layer 0: 6 -> 16 (padded to 8), relations {connected_to, ordered_next}
    launch_zero(h0, (long)NN * 16, stream);
    for (int r = 0; r < 2; ++r) {
        const int* ei = (const int*)d_in[1 + r];
        launch_zero(msg, (long)NN * 8, stream);
        launch_scatter(xpad, ei, msg, 8, stream);      // pad cols scatter zeros
        launch_gemm2<8, 16, false>(msg, w0np + r * 8 * 16,
                                   xpad, w0rp + r * 8 * 16,
                                   b0 + (long)r * 16, h0, NN, stream);
    }

    // ---- layer 1: 16 -> 32, same relations, + h0 @ P1 + pb1, fused relu
    launch_zero(h1, (long)NN * 32, stream);
    for (int r = 0; r < 2; ++r) {
        const int* ei = (const int*)d_in[1 + r];
        launch_zero(msg, (long)NN * 16, stream);
        launch_scatter(h0, ei, msg, 16, stream);
        launch_gemm2<16, 32, false>(msg, W1n + (long)r * 16 * 32,
                                    h0,  W1r + (long)r * 16 * 32,
                                    b1 + (long)r * 32, h1, NN, stream);
    }
    launch_gemm2<16, 32, true>(h0, P1, nullptr, nullptr, pb1, h1, NN, stream);

    // ---- layer 2: 32 -> 64, 7 relations with sum/mean, + h1 @ P2 + pb2, relu
    static const int is_mean[7] = {0, 0, 1, 1, 0, 0, 1}; // represents..perpendicular
    launch_zero(out, (long)NN * 64, stream);
    for (int r = 0; r < 7; ++r) {
        const int* ei = (const int*)d_in[3 + r];
        launch_zero(msg, (long)NN * 32, stream);
        launch_scatter(h1, ei, msg, 32, stream);
        if (is_mean[r]) {
            launch_zero(cnt, NN, stream);
            scatter_count_kernel<<<(NE + 255) / 256, 256, 0, stream>>>(ei + NE, cnt);
            long total = (long)NN * 32;
            mean_div_kernel<<<(int)((total + 255) / 256), 256, 0, stream>>>(msg, cnt, 32);
        }
        launch_gemm2<32, 64, false>(msg, W2n + (long)r * 32 * 64,
                                    h1,  W2r + (long)r * 32 * 64,
                                    b2 + (long)r * 64, out, NN, stream);
    }
    // final projection with fused relu (relu(relu(z)) == relu(z))
    launch_gemm2<32, 64, true>(h1, P2, nullptr, nullptr, pb2, out, NN, stream);
}